// AttentionFlow_20993800142939
// MI455X (gfx1250) — compile-verified
//
#include <hip/hip_runtime.h>
#include <stdint.h>

// ---------------------------------------------------------------------------
// AttentionFlow on MI455X (gfx1250, wave32).
// Strategy: factor logit = left^T (Wq^T Wk) right into block terms so the
// heavy math becomes (rows x 32) @ (32 x 32) fp32 GEMMs executed with
// V_WMMA_F32_16X16X4_F32, plus cheap per-edge 32-dot products (one wave per
// edge, lane = feature dim, shfl_xor reduction). Softmax / top-k / scatter
// use f32 atomics on L2 (node arrays are L2-resident: ~50MB << 192MB).
// ---------------------------------------------------------------------------

typedef __attribute__((ext_vector_type(2))) float v2f;
typedef __attribute__((ext_vector_type(8))) float v8f;

#define DD 32
#define D4 128

__device__ __forceinline__ unsigned mapf(float x) {
  unsigned b = __float_as_uint(x);
  return (b & 0x80000000u) ? ~b : (b | 0x80000000u);
}
__device__ __forceinline__ float unmapf(unsigned u) {
  return (u & 0x80000000u) ? __uint_as_float(u ^ 0x80000000u)
                           : __uint_as_float(~u);
}

// ---------------- zero scratch ----------------
__global__ void k_zero_u32(unsigned* p, long long n) {
  long long i = (long long)blockIdx.x * blockDim.x + threadIdx.x;
  long long stride = (long long)gridDim.x * blockDim.x;
  for (; i < n; i += stride) p[i] = 0u;
}

// ---------------- M = Wq^T @ Wk (128x128) ----------------
__global__ void k_computeM(const float* __restrict__ Wq,
                           const float* __restrict__ Wk,
                           float* __restrict__ M) {
  int idx = blockIdx.x * blockDim.x + threadIdx.x;
  if (idx >= D4 * D4) return;
  int i = idx / D4, j = idx % D4;
  float s = 0.f;
  for (int k = 0; k < D4; ++k) s += Wq[k * D4 + i] * Wk[k * D4 + j];
  M[idx] = s;
}

// ---------------- extract 32x32 block operands ----------------
// BP[d][j]=M00[d][j]  BQ[d][j]=M01[d][j]  BR[d][j]=M10[j][d]
// BT[d][j]=M11[j][d]  BW[d][j]=Wlin[j][d]
__global__ void k_extractB(const float* __restrict__ M,
                           const float* __restrict__ Wlin,
                           float* BP, float* BQ, float* BR, float* BT,
                           float* BW) {
  int idx = blockIdx.x * blockDim.x + threadIdx.x;
  if (idx >= DD * DD) return;
  int d = idx / DD, j = idx % DD;
  BP[idx] = M[d * D4 + j];
  BQ[idx] = M[d * D4 + DD + j];
  BR[idx] = M[(DD + j) * D4 + d];
  BT[idx] = M[(DD + j) * D4 + DD + d];
  BW[idx] = Wlin[j * DD + d];
}

// ---------------- per-query precompute (128 queries) ----------------
__global__ void k_queryPre(const float* __restrict__ M,
                           const float* __restrict__ qse,
                           const float* __restrict__ qre,
                           float* qu, float* qv, float* qwx, float* qc) {
  int q = blockIdx.x;
  __shared__ float z[64];
  __shared__ float part[64];
  int tid = threadIdx.x;
  if (tid < DD) z[tid] = qse[q * DD + tid];
  else          z[tid] = qre[q * DD + (tid - DD)];
  __syncthreads();
  if (tid < DD) {
    int j = tid;
    float u = 0.f, v = 0.f, wv = 0.f, x = 0.f;
    for (int d = 0; d < DD; ++d) {
      float qd = z[d], rd = z[DD + d];
      u  += M[j * D4 + 2 * DD + d] * qd + M[j * D4 + 3 * DD + d] * rd;
      v  += M[(2 * DD + d) * D4 + j] * qd + M[(3 * DD + d) * D4 + j] * rd;
      wv += M[(DD + j) * D4 + 2 * DD + d] * qd +
            M[(DD + j) * D4 + 3 * DD + d] * rd;
      x  += M[(2 * DD + d) * D4 + DD + j] * qd +
            M[(3 * DD + d) * D4 + DD + j] * rd;
    }
    qu[q * DD + j] = u;
    qv[q * DD + j] = v;
    qwx[q * DD + j] = wv + x;
  }
  // c = z^T M[64:128,64:128] z
  float p = 0.f;
  {
    int i = tid;
    float zi = z[i];
    for (int j2 = 0; j2 < 64; ++j2)
      p += zi * M[(2 * DD + i) * D4 + 2 * DD + j2] * z[j2];
  }
  part[tid] = p;
  __syncthreads();
  for (int s = 32; s > 0; s >>= 1) {
    if (tid < s) part[tid] += part[tid + s];
    __syncthreads();
  }
  if (tid == 0) qc[q] = part[0];
}

// ---------------- WMMA fp32 GEMM: C[rows x 32] = A[rows x 32] @ B[32 x 32] --
// One wave per 16-row tile, two 16-col accumulators, K=32 in 8 steps of 4
// using V_WMMA_F32_16X16X4_F32. Optional bias + LeakyReLU epilogue.
__global__ void k_gemm32(const float* __restrict__ A,
                         const float* __restrict__ B,
                         float* __restrict__ C, int rows,
                         const float* __restrict__ bias, int act) {
  int wavesPerBlock = blockDim.x >> 5;
  int wave = blockIdx.x * wavesPerBlock + (threadIdx.x >> 5);
  int row0 = wave * 16;
  if (row0 >= rows) return;  // wave-uniform exit (EXEC stays all-ones)
  int lane = threadIdx.x & 31;
  int half = lane >> 4;   // 0: K pair {k,k+1}, 1: {k+2,k+3}
  int l = lane & 15;      // row (A) / col (B)

  v8f acc0 = {0.f, 0.f, 0.f, 0.f, 0.f, 0.f, 0.f, 0.f};
  v8f acc1 = {0.f, 0.f, 0.f, 0.f, 0.f, 0.f, 0.f, 0.f};
  const float* Arow = A + (size_t)(row0 + l) * DD;
#pragma unroll
  for (int kk = 0; kk < DD; kk += 4) {
    int k = kk + 2 * half;
    v2f a;  a.x = Arow[k];               a.y = Arow[k + 1];
    v2f b0; b0.x = B[k * DD + l];        b0.y = B[(k + 1) * DD + l];
    v2f b1; b1.x = B[k * DD + 16 + l];   b1.y = B[(k + 1) * DD + 16 + l];
    acc0 = __builtin_amdgcn_wmma_f32_16x16x4_f32(false, a, false, b0,
                                                 (short)0, acc0, false, false);
    acc1 = __builtin_amdgcn_wmma_f32_16x16x4_f32(false, a, false, b1,
                                                 (short)0, acc1, false, false);
  }
#pragma unroll
  for (int i = 0; i < 8; ++i) {
    int r = row0 + i + 8 * half;   // C layout: VGPR i -> row i / i+8 per half
    if (r < rows) {
      float v0 = acc0[i], v1 = acc1[i];
      if (bias) { v0 += bias[l]; v1 += bias[16 + l]; }
      if (act) {
        v0 = v0 >= 0.f ? v0 : 0.01f * v0;
        v1 = v1 >= 0.f ? v1 : 0.01f * v1;
      }
      C[(size_t)r * DD + l] = v0;
      C[(size_t)r * DD + 16 + l] = v1;
    }
  }
}

// ---------------- per-edge logits: one wave per edge ----------------
__global__ void k_edgeLogits(int E, const int* __restrict__ src,
                             const int* __restrict__ dst,
                             const int* __restrict__ eg,
                             const float* __restrict__ rel,
                             const float* __restrict__ t,
                             const float* __restrict__ repr,
                             const float* __restrict__ P,
                             const float* __restrict__ Qm,
                             const float* __restrict__ R,
                             const float* __restrict__ qu,
                             const float* __restrict__ qv,
                             const float* __restrict__ qwx,
                             const float* __restrict__ qc,
                             float* __restrict__ elog) {
  int wavesPerBlock = blockDim.x >> 5;
  int wave = blockIdx.x * wavesPerBlock + (threadIdx.x >> 5);
  int nWaves = gridDim.x * wavesPerBlock;
  int d = threadIdx.x & 31;
  for (int e = wave; e < E; e += nWaves) {
    int s = src[e], dn = dst[e], g = eg[e];
    size_t so = (size_t)s * DD + d;
    size_t doo = (size_t)dn * DD + d;
    size_t eo = (size_t)e * DD + d;
    size_t go = (size_t)g * DD + d;
    float rs = repr[so], rd = repr[doo];
    float re = rel[eo], te = t[eo];
    float p = P[so] * rd + Qm[so] * re + re * R[doo] + rs * qu[go] +
              rd * qv[go] + re * (qwx[go] + te);
    for (int m = 16; m >= 1; m >>= 1) p += __shfl_xor(p, m, 32);
    if (d == 0) elog[e] = p + qc[g];
  }
}

// ---------------- segment softmax over src node ----------------
__global__ void k_segMax(int E, const float* __restrict__ elog,
                         const int* __restrict__ src, unsigned* nodeMaxU) {
  long long i = (long long)blockIdx.x * blockDim.x + threadIdx.x;
  long long stride = (long long)gridDim.x * blockDim.x;
  for (; i < E; i += stride)
    atomicMax(&nodeMaxU[src[i]], mapf(elog[i]));
}
__global__ void k_segExp(int E, const float* __restrict__ elog,
                         const int* __restrict__ src,
                         const unsigned* __restrict__ nodeMaxU,
                         float* nodeSum, float* eex) {
  long long i = (long long)blockIdx.x * blockDim.x + threadIdx.x;
  long long stride = (long long)gridDim.x * blockDim.x;
  for (; i < E; i += stride) {
    int s = src[i];
    float ex = __expf(elog[i] - unmapf(nodeMaxU[s]));
    eex[i] = ex;
    atomicAdd(&nodeSum[s], ex);
  }
}
__global__ void k_segNorm(int E, float* __restrict__ w,
                          const int* __restrict__ src,
                          const float* __restrict__ nodeSum,
                          float* __restrict__ tgt,
                          const float* __restrict__ score) {
  long long i = (long long)blockIdx.x * blockDim.x + threadIdx.x;
  long long stride = (long long)gridDim.x * blockDim.x;
  for (; i < E; i += stride) {
    int s = src[i];
    float v = w[i] / nodeSum[s];
    w[i] = v;
    if (tgt) tgt[i] = v * score[s];
  }
}

// ---------------- segment starts for sorted eg ----------------
__global__ void k_segStart(int E, int Qn, const int* __restrict__ eg,
                           int* segStart) {
  int q = blockIdx.x * blockDim.x + threadIdx.x;
  if (q > Qn) return;
  if (q == Qn) { segStart[q] = E; return; }
  int lo = 0, hi = E;
  while (lo < hi) {
    int mid = (lo + hi) >> 1;
    if (eg[mid] < q) lo = mid + 1; else hi = mid;
  }
  segStart[q] = lo;
}

// ---------------- per-query top-K prune (radix binary search) -------------
// One block per query segment. Finds the K-th largest mapped key, keeps
// values > thr plus enough == thr entries, zeroes w for the rest.
__global__ void k_topk(const float* __restrict__ tgt, float* __restrict__ w,
                       const int* __restrict__ segStart,
                       const int* __restrict__ kPtr) {
  int q = blockIdx.x;
  int s = segStart[q], epos = segStart[q + 1];
  int cnt = epos - s;
  int K = *kPtr;
  if (cnt <= K) return;
  __shared__ int red[256];
  __shared__ int sh_needEq;
  __shared__ int sh_eq;
  int tid = threadIdx.x;
  unsigned p = 0;
  for (int bit = 31; bit >= 0; --bit) {
    unsigned cand = p | (1u << bit);
    int local = 0;
    for (int i = s + tid; i < epos; i += blockDim.x)
      local += (mapf(tgt[i]) >= cand) ? 1 : 0;
    red[tid] = local;
    __syncthreads();
    for (int st = 128; st > 0; st >>= 1) {
      if (tid < st) red[tid] += red[tid + st];
      __syncthreads();
    }
    int total = red[0];
    __syncthreads();
    if (total >= K) p = cand;
  }
  {
    int local = 0;
    for (int i = s + tid; i < epos; i += blockDim.x)
      local += (mapf(tgt[i]) > p) ? 1 : 0;
    red[tid] = local;
    __syncthreads();
    for (int st = 128; st > 0; st >>= 1) {
      if (tid < st) red[tid] += red[tid + st];
      __syncthreads();
    }
    if (tid == 0) { sh_needEq = K - red[0]; sh_eq = 0; }
    __syncthreads();
  }
  for (int i = s + tid; i < epos; i += blockDim.x) {
    unsigned m = mapf(tgt[i]);
    bool keep;
    if (m > p) keep = true;
    else if (m == p) keep = (atomicAdd(&sh_eq, 1) < sh_needEq);  // ties rare
    else keep = false;
    if (!keep) w[i] = 0.f;
  }
}

// ---------------- scatter aggregation: one wave per edge ----------------
__global__ void k_agg(int E, const float* __restrict__ w,
                      const int* __restrict__ src, const int* __restrict__ dst,
                      const float* __restrict__ repr, float* __restrict__ agg,
                      float* __restrict__ scoreOut,
                      const float* __restrict__ score) {
  int wavesPerBlock = blockDim.x >> 5;
  int wave = blockIdx.x * wavesPerBlock + (threadIdx.x >> 5);
  int nWaves = gridDim.x * wavesPerBlock;
  int d = threadIdx.x & 31;
  for (int e = wave; e < E; e += nWaves) {
    float we = w[e];
    if (we != 0.f) {
      int s = src[e], dn = dst[e];
      atomicAdd(&agg[(size_t)dn * DD + d], we * repr[(size_t)s * DD + d]);
      if (scoreOut && d == 0) atomicAdd(&scoreOut[dn], we * score[s]);
    }
  }
}

// ---------------- blend: out = 0.8*agg + 0.2*prev ----------------
__global__ void k_blend(long long n, const float* __restrict__ agg,
                        const float* __restrict__ prev,
                        float* __restrict__ out) {
  long long i = (long long)blockIdx.x * blockDim.x + threadIdx.x;
  long long stride = (long long)gridDim.x * blockDim.x;
  for (; i < n; i += stride) out[i] = 0.8f * agg[i] + 0.2f * prev[i];
}

// ---------------------------------------------------------------------------
extern "C" void kernel_launch(void* const* d_in, const int* in_sizes, int n_in,
                              void* d_out, int out_size, void* d_ws,
                              size_t ws_size, hipStream_t stream) {
  const float* score     = (const float*)d_in[0];
  const float* node_repr = (const float*)d_in[1];
  const int*   eg0  = (const int*)d_in[2];
  const int*   src0 = (const int*)d_in[3];
  const int*   dst0 = (const int*)d_in[4];
  const float* rel0 = (const float*)d_in[5];
  const int*   eg1  = (const int*)d_in[6];
  const int*   src1 = (const int*)d_in[7];
  const int*   dst1 = (const int*)d_in[8];
  const float* rel1 = (const float*)d_in[9];
  const float* qse  = (const float*)d_in[10];
  const float* qre  = (const float*)d_in[11];
  const float* Wq   = (const float*)d_in[12];
  const float* Wk   = (const float*)d_in[13];
  const float* Wlin = (const float*)d_in[14];
  const float* blin = (const float*)d_in[15];
  const int*   kPtr = (const int*)d_in[16];

  const int N  = in_sizes[0];
  const int E  = in_sizes[2];
  const int Qn = in_sizes[10] / DD;

  // workspace layout (floats)
  float* w = (float*)d_ws;
  size_t o = 0;
  float* M  = w + o; o += (size_t)D4 * D4;
  float* BP = w + o; o += DD * DD;
  float* BQ = w + o; o += DD * DD;
  float* BR = w + o; o += DD * DD;
  float* BT = w + o; o += DD * DD;
  float* BW = w + o; o += DD * DD;
  float* qu  = w + o; o += (size_t)Qn * DD;
  float* qv  = w + o; o += (size_t)Qn * DD;
  float* qwx = w + o; o += (size_t)Qn * DD;
  float* qc  = w + o; o += Qn;
  float* P  = w + o; o += (size_t)N * DD;
  float* Qm = w + o; o += (size_t)N * DD;
  float* R  = w + o; o += (size_t)N * DD;
  float* reprA = w + o; o += (size_t)N * DD;
  float* reprB = w + o; o += (size_t)N * DD;
  float* aggB  = w + o; o += (size_t)N * DD;
  float* t    = w + o; o += (size_t)E * DD;
  float* elog = w + o; o += E;
  float* eex  = w + o; o += E;
  float* tgt  = w + o; o += E;
  unsigned* nodeMaxU = (unsigned*)(w + o); o += N;
  float* nodeSum = w + o; o += N;
  int* segS = (int*)(w + o); o += Qn + 1;

  float* scoreOut = (float*)d_out;
  float* reprOut  = (float*)d_out + N;

  const int TB = 256;
  const int gemmBlocksN = (N + 8 * 16 - 1) / (8 * 16);  // 8 waves/block
  const int gemmBlocksE = (E + 8 * 16 - 1) / (8 * 16);
  const int edgeBlocks = 4096;                          // wave-per-edge loops
  const int edgeTB = (E + TB - 1) / TB;

  // ---- shared preprocessing ----
  k_computeM<<<(D4 * D4 + TB - 1) / TB, TB, 0, stream>>>(Wq, Wk, M);
  k_extractB<<<(DD * DD + TB - 1) / TB, TB, 0, stream>>>(M, Wlin, BP, BQ, BR,
                                                         BT, BW);
  k_queryPre<<<Qn, 64, 0, stream>>>(M, qse, qre, qu, qv, qwx, qc);

  // ---- zero everything we accumulate into for layer 1 ----
  k_zero_u32<<<2048, TB, 0, stream>>>(nodeMaxU, N);
  k_zero_u32<<<2048, TB, 0, stream>>>((unsigned*)nodeSum, N);
  k_zero_u32<<<4096, TB, 0, stream>>>((unsigned*)aggB, (long long)N * DD);
  k_zero_u32<<<2048, TB, 0, stream>>>((unsigned*)scoreOut, N);

  // ======== layer 1 (uses node_repr, with top-K prune) ========
  k_gemm32<<<gemmBlocksN, TB, 0, stream>>>(node_repr, BP, P, N, nullptr, 0);
  k_gemm32<<<gemmBlocksN, TB, 0, stream>>>(node_repr, BQ, Qm, N, nullptr, 0);
  k_gemm32<<<gemmBlocksN, TB, 0, stream>>>(node_repr, BR, R, N, nullptr, 0);
  k_gemm32<<<gemmBlocksE, TB, 0, stream>>>(rel1, BT, t, E, nullptr, 0);
  k_edgeLogits<<<edgeBlocks, TB, 0, stream>>>(E, src1, dst1, eg1, rel1, t,
                                              node_repr, P, Qm, R, qu, qv,
                                              qwx, qc, elog);
  k_segMax<<<edgeTB, TB, 0, stream>>>(E, elog, src1, nodeMaxU);
  k_segExp<<<edgeTB, TB, 0, stream>>>(E, elog, src1, nodeMaxU, nodeSum, eex);
  k_segNorm<<<edgeTB, TB, 0, stream>>>(E, eex, src1, nodeSum, tgt, score);
  k_segStart<<<(Qn + 1 + TB - 1) / TB, TB, 0, stream>>>(E, Qn, eg1, segS);
  k_topk<<<Qn, TB, 0, stream>>>(tgt, eex, segS, kPtr);
  k_agg<<<edgeBlocks, TB, 0, stream>>>(E, eex, src1, dst1, node_repr, aggB,
                                       scoreOut, score);
  k_blend<<<4096, TB, 0, stream>>>((long long)N * DD, aggB, node_repr, reprA);

  // ---- re-zero accumulators for layer 0 ----
  k_zero_u32<<<2048, TB, 0, stream>>>(nodeMaxU, N);
  k_zero_u32<<<2048, TB, 0, stream>>>((unsigned*)nodeSum, N);
  k_zero_u32<<<4096, TB, 0, stream>>>((unsigned*)aggB, (long long)N * DD);

  // ======== layer 0 (uses reprA, no prune) ========
  k_gemm32<<<gemmBlocksN, TB, 0, stream>>>(reprA, BP, P, N, nullptr, 0);
  k_gemm32<<<gemmBlocksN, TB, 0, stream>>>(reprA, BQ, Qm, N, nullptr, 0);
  k_gemm32<<<gemmBlocksN, TB, 0, stream>>>(reprA, BR, R, N, nullptr, 0);
  k_gemm32<<<gemmBlocksE, TB, 0, stream>>>(rel0, BT, t, E, nullptr, 0);
  k_edgeLogits<<<edgeBlocks, TB, 0, stream>>>(E, src0, dst0, eg0, rel0, t,
                                              reprA, P, Qm, R, qu, qv, qwx,
                                              qc, elog);
  k_segMax<<<edgeTB, TB, 0, stream>>>(E, elog, src0, nodeMaxU);
  k_segExp<<<edgeTB, TB, 0, stream>>>(E, elog, src0, nodeMaxU, nodeSum, eex);
  k_segNorm<<<edgeTB, TB, 0, stream>>>(E, eex, src0, nodeSum, nullptr,
                                       nullptr);
  k_agg<<<edgeBlocks, TB, 0, stream>>>(E, eex, src0, dst0, reprA, aggB,
                                       nullptr, nullptr);
  k_blend<<<4096, TB, 0, stream>>>((long long)N * DD, aggB, reprA, reprB);

  // ======== bypass: LeakyReLU(reprB @ Wlin^T + blin) -> d_out[N:] ========
  k_gemm32<<<gemmBlocksN, TB, 0, stream>>>(reprB, BW, reprOut, N, blin, 1);
}